// PrototypeExtractor_46196668236382
// MI455X (gfx1250) — compile-verified
//
#include <hip/hip_runtime.h>
#include <math.h>

// Problem dims (from reference setup_inputs)
#define K_SUP 8
#define C_DIM 256
#define HW    16384   // 128*128
#define B_Q   8
#define NCLS  3
#define EPS_NORM 1e-12f

typedef float v2f __attribute__((ext_vector_type(2)));
typedef float v8f __attribute__((ext_vector_type(8)));

// Workspace layout (floats):
//   [0,    768)  numer[3][256]        (zeroed per launch)
//   [768,  771)  counts[3]            (zeroed per launch)
//   [1024, 3072) bmat0[64][32]        (B-matrix VGPR0 lane image, rewritten per launch)
//   [3072, 5120) bmat1[64][32]        (B-matrix VGPR1 lane image, rewritten per launch)

// ---------------------------------------------------------------------------
// Kernel 1: masked per-class channel sums + class pixel counts.
// One thread per channel; each block handles (k, 1024-pixel chunk).
// Per-thread contiguous float4 streaming; 3 atomics per thread at the end.
// ---------------------------------------------------------------------------
__global__ void proto_accum(const float* __restrict__ sf,
                            const int*   __restrict__ sm,
                            float* __restrict__ numer,
                            float* __restrict__ counts) {
  const int c  = threadIdx.x;          // 0..255 channel
  const int k  = blockIdx.x;           // 0..7 support image
  const int p0 = blockIdx.y * 1024;    // pixel chunk base
  const float* f = sf + ((size_t)(k * C_DIM + c)) * HW + p0;
  const int*   m = sm + (size_t)k * HW + p0;

  float a0 = 0.f, a1 = 0.f, a2 = 0.f;
  int   n0 = 0,   n1 = 0,   n2 = 0;
  const bool counter = (threadIdx.x == 0);

  for (int p = 0; p < 1024; p += 4) {
    const int4   mm = *(const int4*)(m + p);
    const float4 ff = *(const float4*)(f + p);
    a0 += (mm.x == 0) ? ff.x : 0.f;  a1 += (mm.x == 1) ? ff.x : 0.f;  a2 += (mm.x == 2) ? ff.x : 0.f;
    a0 += (mm.y == 0) ? ff.y : 0.f;  a1 += (mm.y == 1) ? ff.y : 0.f;  a2 += (mm.y == 2) ? ff.y : 0.f;
    a0 += (mm.z == 0) ? ff.z : 0.f;  a1 += (mm.z == 1) ? ff.z : 0.f;  a2 += (mm.z == 2) ? ff.z : 0.f;
    a0 += (mm.w == 0) ? ff.w : 0.f;  a1 += (mm.w == 1) ? ff.w : 0.f;  a2 += (mm.w == 2) ? ff.w : 0.f;
    if (counter) {
      n0 += (mm.x == 0) + (mm.y == 0) + (mm.z == 0) + (mm.w == 0);
      n1 += (mm.x == 1) + (mm.y == 1) + (mm.z == 1) + (mm.w == 1);
      n2 += (mm.x == 2) + (mm.y == 2) + (mm.z == 2) + (mm.w == 2);
    }
  }
  atomicAdd(&numer[0 * C_DIM + c], a0);
  atomicAdd(&numer[1 * C_DIM + c], a1);
  atomicAdd(&numer[2 * C_DIM + c], a2);
  if (counter) {
    atomicAdd(&counts[0], (float)n0);
    atomicAdd(&counts[1], (float)n1);
    atomicAdd(&counts[2], (float)n2);
  }
}

// ---------------------------------------------------------------------------
// Kernel 2: finalize prototypes (divide, normalize twice), emit:
//   - prototypes (after first normalize) to d_out tail
//   - twice-normalized prototypes packed into the WMMA B-matrix lane layout:
//     B is 4(K)x16(N); VGPR0: lanes0-15 row k0, lanes16-31 row k0+2;
//     VGPR1: rows k0+1 / k0+3. Columns 0..2 = classes, 3..15 = zero.
// ---------------------------------------------------------------------------
__global__ void proto_finalize(const float* __restrict__ numer,
                               const float* __restrict__ counts,
                               float* __restrict__ proto_out,
                               float* __restrict__ bmat0,
                               float* __restrict__ bmat1) {
  __shared__ float red[256];
  __shared__ float p2s[NCLS * C_DIM];
  const int tid = threadIdx.x;

  for (int n = 0; n < NCLS; ++n) {
    float v = numer[n * C_DIM + tid] / fmaxf(counts[n], 1.0f);
    // first L2 norm
    red[tid] = v * v; __syncthreads();
    for (int s = 128; s > 0; s >>= 1) { if (tid < s) red[tid] += red[tid + s]; __syncthreads(); }
    const float p1 = v / fmaxf(sqrtf(red[0]), EPS_NORM);
    __syncthreads();
    proto_out[n * C_DIM + tid] = p1;
    // second L2 norm (reference normalizes again)
    red[tid] = p1 * p1; __syncthreads();
    for (int s = 128; s > 0; s >>= 1) { if (tid < s) red[tid] += red[tid + s]; __syncthreads(); }
    const float p2 = p1 / fmaxf(sqrtf(red[0]), EPS_NORM);
    __syncthreads();
    p2s[n * C_DIM + tid] = p2;
  }
  __syncthreads();

  // Pack B-matrix lane images: entry e = chunk*32 + lane
  for (int e = tid; e < 64 * 32; e += 256) {
    const int chunk = e >> 5;
    const int lane  = e & 31;
    const int n     = lane & 15;
    const int kk    = chunk * 4 + ((lane >> 4) << 1);  // k0 or k0+2
    float b0 = 0.f, b1 = 0.f;
    if (n < NCLS) { b0 = p2s[n * C_DIM + kk]; b1 = p2s[n * C_DIM + kk + 1]; }
    bmat0[e] = b0;
    bmat1[e] = b1;
  }
}

// ---------------------------------------------------------------------------
// Kernel 3: similarity via V_WMMA_F32_16X16X4_F32.
// One wave per 16-pixel tile: A = q[16 pixels x 4 channels], B = prototypes,
// K = 256 in 64 chunks. ||q||^2 accumulated in VALU from the same A loads,
// combined across lane halves; epilogue folds both normalize-q and /temp.
// ---------------------------------------------------------------------------
__global__ void sim_wmma(const float* __restrict__ q,
                         const float* __restrict__ bmat0,
                         const float* __restrict__ bmat1,
                         const float* __restrict__ temp,
                         float* __restrict__ sim) {
  const int lane  = threadIdx.x & 31;
  const int wave  = threadIdx.x >> 5;
  const int tile  = blockIdx.x * 8 + wave;     // 8192 tiles total
  const int b     = tile >> 10;                // 1024 tiles per batch image
  const int p0    = (tile & 1023) << 4;        // pixel base
  const int mrow  = lane & 15;                 // A row (pixel within tile)
  const int khalf = (lane >> 4) << 1;          // 0 (lanes 0-15) or 2 (lanes 16-31)

  const float* qb = q + (size_t)b * C_DIM * HW + p0 + mrow;

  v8f  acc = {};
  float qq = 0.f;

#pragma unroll 4
  for (int chunk = 0; chunk < 64; ++chunk) {
    const int k0 = chunk * 4 + khalf;
    const float ax = qb[(size_t)k0 * HW];
    const float ay = qb[(size_t)(k0 + 1) * HW];
    v2f A;  A.x = ax;              A.y = ay;
    v2f Bv; Bv.x = bmat0[chunk * 32 + lane]; Bv.y = bmat1[chunk * 32 + lane];
    acc = __builtin_amdgcn_wmma_f32_16x16x4_f32(false, A, false, Bv,
                                                (short)0, acc, false, false);
    qq = __builtin_fmaf(ax, ax, qq);
    qq = __builtin_fmaf(ay, ay, qq);
  }

  // combine the two lane-half partial ||q||^2 (pixel m lives in lanes m and m+16)
  qq += __shfl_xor(qq, 16, 32);

  const float tdiv  = fmaxf(temp[0], 0.01f);
  const int   ncol  = lane & 15;               // D column = class id
  const int   mbase = (lane >> 4) << 3;        // D rows: i (+8 for upper lanes)
  float* ob = sim + ((size_t)(b * NCLS + ncol)) * HW + p0;

#pragma unroll
  for (int i = 0; i < 8; ++i) {
    const float qv  = __shfl(qq, mbase + i, 32);       // ||q||^2 of this row's pixel
    const float val = acc[i] / (fmaxf(sqrtf(qv), EPS_NORM) * tdiv);
    if (ncol < NCLS) ob[mbase + i] = val;
  }
}

// ---------------------------------------------------------------------------
extern "C" void kernel_launch(void* const* d_in, const int* in_sizes, int n_in,
                              void* d_out, int out_size, void* d_ws, size_t ws_size,
                              hipStream_t stream) {
  const float* sf   = (const float*)d_in[0];   // [8,256,128,128]
  const int*   sm   = (const int*)  d_in[1];   // [8,128,128]
  const float* qf   = (const float*)d_in[2];   // [8,256,128,128]
  const float* temp = (const float*)d_in[3];   // [1]

  float* out   = (float*)d_out;                // sim [8,3,128,128] then proto [3,256]
  float* ws    = (float*)d_ws;
  float* numer  = ws;                          // 768
  float* counts = ws + 768;                    // 3
  float* bmat0  = ws + 1024;                   // 2048
  float* bmat1  = ws + 3072;                   // 2048

  // zero the accumulators (memset node is graph-capture legal)
  hipMemsetAsync(d_ws, 0, 1024 * sizeof(float), stream);

  proto_accum   <<<dim3(K_SUP, 16), 256, 0, stream>>>(sf, sm, numer, counts);
  proto_finalize<<<1, 256, 0, stream>>>(numer, counts,
                                        out + (size_t)B_Q * NCLS * HW, bmat0, bmat1);
  sim_wmma      <<<1024, 256, 0, stream>>>(qf, bmat0, bmat1, temp, out);
}